// ExpertChoiceRouter_29429115912369
// MI455X (gfx1250) — compile-verified
//
#include <hip/hip_runtime.h>
#include <hip/hip_bf16.h>

typedef __attribute__((ext_vector_type(2))) float v2f;
typedef __attribute__((ext_vector_type(4))) float v4f;
typedef __attribute__((ext_vector_type(8))) float v8f;

#define B_      8
#define S_      4096
#define H_      1024
#define K_SEL   2048u

// ---------------- Kernel 1: router GEMV via V_WMMA_F32_16X16X4_F32 ----------
// grid = (B*S)/128 blocks of 256 threads (8 waves). Each wave computes the
// 1024-length dot product for 16 tokens, accumulating with WMMA f32 16x16x4:
//   A[m][k] = hidden[token m][kbase+k]   (16x4 tile)
//   B[k][n] = w[kbase+k]                 (broadcast across N)
// => D[m][n] = logit(token m) for every n; no cross-lane reduction needed.
// A layout (ISA 7.12.2): lanes 0-15 own K={0,1}, lanes 16-31 own K={2,3}.

#define WAVES_PB   8
#define TOK_PW     16
#define TOK_PB     (WAVES_PB * TOK_PW)   // 128 tokens per block
#define KTILE      32
#define ROWPAD     36                    // bank spread 4*(9t mod 16), 16B-aligned rows

__global__ __launch_bounds__(256) void router_gemv_wmma(
    const float* __restrict__ hs, const float* __restrict__ w,
    float* __restrict__ scores)
{
    __shared__ float s_w[H_];
    __shared__ float s_tile[WAVES_PB][TOK_PW * ROWPAD];

    const int tid = threadIdx.x;
    ((v4f*)s_w)[tid] = ((const v4f*)w)[tid];     // 256 * float4 = 1024 floats
    __syncthreads();

    const int wave = tid >> 5;
    const int lane = tid & 31;
    const int tok0 = blockIdx.x * TOK_PB + wave * TOK_PW;

    // staging: lane pair (2t, 2t+1) covers 128 contiguous bytes of row t
    const int t = lane >> 1;
    const int h = lane & 1;
    const float* rowp = hs + (size_t)(tok0 + t) * H_ + h * 16;
    float* tp = &s_tile[wave][t * ROWPAD + h * 16];

    const int khalf  = (lane < 16) ? 0 : 2;   // K pair owned by this lane half
    const int tokrow = lane & 15;             // M index of the A fragment

    v8f c = {};  // D[m][n] accumulator (logits, identical across n)

    for (int kt = 0; kt < H_ / KTILE; ++kt) {
        const float* gp = rowp + kt * KTILE;
        __builtin_prefetch(gp + KTILE, 0, 1);        // global_prefetch_b8
        v4f r0 = *(const v4f*)(gp + 0);
        v4f r1 = *(const v4f*)(gp + 4);
        v4f r2 = *(const v4f*)(gp + 8);
        v4f r3 = *(const v4f*)(gp + 12);
        *(v4f*)(tp + 0)  = r0;
        *(v4f*)(tp + 4)  = r1;
        *(v4f*)(tp + 8)  = r2;
        *(v4f*)(tp + 12) = r3;
        // tile private to this wave; DS ops are in-order within a wave.
        __builtin_amdgcn_wave_barrier();

        #pragma unroll
        for (int kc = 0; kc < KTILE / 4; ++kc) {
            const int kk = kc * 4 + khalf;
            v2f a = *(const v2f*)&s_tile[wave][tokrow * ROWPAD + kk]; // ds_load_b64
            v2f b = *(const v2f*)&s_w[kt * KTILE + kk];               // ds_load_b64
            c = __builtin_amdgcn_wmma_f32_16x16x4_f32(
                    false, a, false, b, (short)0, c, false, false);
        }
        __builtin_amdgcn_wave_barrier();
    }

    // C/D layout: vgpr r, lanes 0-15 -> (M=r, N=lane); lanes 16-31 -> (M=r+8).
    if (lane == 0 || lane == 16) {
        const int mbase = (lane == 0) ? 0 : 8;
        #pragma unroll
        for (int r = 0; r < 8; ++r) {
            float sc = 1.0f / (1.0f + __expf(-c[r]));  // sigmoid(alpha=1)
            scores[tok0 + mbase + r] = sc;
        }
    }
}

// ---------------- Kernel 2: exact per-row top-K, 8-bit radix histogram ------
// One workgroup (1024 threads) per batch row. Scores are strictly positive so
// uint bit order == float order. 4 MSB-first 8-bit digit passes with 8-way
// privatized 256-bin LDS histograms; wave 0 picks each digit with shfl suffix
// scans. After the last pass, prefix == exact K-th largest key and the running
// K' equals the number of tie elements to admit (index order, as jax top_k).
#define NHIST 8

__global__ __launch_bounds__(1024) void topk_select(
    const float* __restrict__ scores, float* __restrict__ out)
{
    __shared__ float    s_sc[S_];
    __shared__ unsigned s_hist[NHIST][256];
    __shared__ unsigned s_wsum[32];
    __shared__ unsigned s_dig;
    __shared__ unsigned s_kp;

    const int b    = blockIdx.x;
    const int tid  = threadIdx.x;
    const int lane = tid & 31;
    const int wv   = tid >> 5;

    ((v4f*)s_sc)[tid] = ((const v4f*)(scores + (size_t)b * S_))[tid];
    if (tid == 0) s_kp = K_SEL;
    __syncthreads();

    unsigned prefix = 0;
    for (int d = 3; d >= 0; --d) {
        for (int i = tid; i < NHIST * 256; i += 1024) ((unsigned*)s_hist)[i] = 0u;
        __syncthreads();

        unsigned* hist = s_hist[wv & (NHIST - 1)];
        const unsigned hs = (unsigned)(d + 1) * 8u;
        #pragma unroll
        for (int j = 0; j < 4; ++j) {
            const unsigned key  = __float_as_uint(s_sc[tid * 4 + j]);
            const bool     cand = (d == 3) || ((key >> hs) == (prefix >> hs));
            if (cand) atomicAdd(&hist[(key >> (d * 8)) & 0xFFu], 1u);
        }
        __syncthreads();

        if (wv == 0) {
            // lane owns digits [8*lane, 8*lane+7]; fold privatized copies
            unsigned hloc[8]; unsigned p = 0;
            #pragma unroll
            for (int j = 0; j < 8; ++j) {
                unsigned s = 0;
                #pragma unroll
                for (int cp = 0; cp < NHIST; ++cp) s += s_hist[cp][lane * 8 + j];
                hloc[j] = s; p += s;
            }
            // inclusive suffix sum over lanes (sum of p for lanes >= mine)
            unsigned suf = p;
            #pragma unroll
            for (int off = 1; off < 32; off <<= 1) {
                unsigned n = __shfl_down(suf, off);
                if (lane + off < 32) suf += n;
            }
            const unsigned above = suf - p;   // count with digit in a higher lane-chunk
            const unsigned kp = s_kp;
            if (above < kp && above + p >= kp) {   // crossing is inside my chunk
                unsigned c = above; bool done = false;
                #pragma unroll
                for (int j = 7; j >= 0; --j) {
                    const unsigned ci = c + hloc[j];
                    if (!done && ci >= kp) {       // first crossing, descending
                        s_dig = (unsigned)(lane * 8 + j);
                        s_kp  = kp - c;            // kp minus strictly-greater
                        done  = true;
                    }
                    c = ci;
                }
            }
        }
        __syncthreads();
        prefix |= s_dig << (d * 8);
        __syncthreads();                           // s_dig/s_hist reused next pass
    }

    const unsigned T     = prefix;   // exact K-th largest key
    const unsigned extra = s_kp;     // #ties to admit, in index order (>=1)

    // index-ordered ranks of tie elements: contiguous 4-chunks + block scan
    unsigned key[4]; unsigned eqf[4]; unsigned eq = 0;
    #pragma unroll
    for (int j = 0; j < 4; ++j) {
        key[j] = __float_as_uint(s_sc[tid * 4 + j]);
        eqf[j] = (key[j] == T) ? 1u : 0u;
        eq += eqf[j];
    }
    unsigned inc = eq;                              // wave inclusive scan
    #pragma unroll
    for (int off = 1; off < 32; off <<= 1) {
        unsigned n = __shfl_up(inc, off);
        if (lane >= off) inc += n;
    }
    if (lane == 31) s_wsum[wv] = inc;
    __syncthreads();
    if (wv == 0) {
        unsigned x = s_wsum[lane];
        #pragma unroll
        for (int off = 1; off < 32; off <<= 1) {
            unsigned n = __shfl_up(x, off);
            if (lane >= off) x += n;
        }
        s_wsum[lane] = x;
    }
    __syncthreads();
    unsigned rank = inc - eq + (wv ? s_wsum[wv - 1] : 0u);  // exclusive, global

    v4f wvec, mvec;
    #pragma unroll
    for (int j = 0; j < 4; ++j) {
        const bool sel = (key[j] > T) || (eqf[j] && rank < extra);
        rank += eqf[j];
        wvec[j] = sel ? s_sc[tid * 4 + j] : 0.0f;
        mvec[j] = sel ? 1.0f : 0.0f;
    }
    ((v4f*)(out + (size_t)b * S_))[tid] = wvec;                       // weights
    ((v4f*)(out + (size_t)(B_ * S_) + (size_t)b * S_))[tid] = mvec;   // mask
}

// ----------------------------------------------------------------------------
extern "C" void kernel_launch(void* const* d_in, const int* in_sizes, int n_in,
                              void* d_out, int out_size, void* d_ws, size_t ws_size,
                              hipStream_t stream) {
    (void)in_sizes; (void)n_in; (void)out_size; (void)ws_size;
    const float* hs = (const float*)d_in[0];   // [B,S,H] fp32
    const float* w  = (const float*)d_in[1];   // [H]     fp32
    float* out      = (float*)d_out;           // [B*S] weights ++ [B*S] mask
    float* scr      = (float*)d_ws;            // [B*S] fp32 scratch (128 KB)

    router_gemv_wmma<<<dim3((B_ * S_) / TOK_PB), dim3(256), 0, stream>>>(hs, w, scr);
    topk_select<<<dim3(B_), dim3(1024), 0, stream>>>(scr, out);
}